// CoarsenBlock_14705968021524
// MI455X (gfx1250) — compile-verified
//
#include <hip/hip_runtime.h>
#include <stdint.h>

// MI455X / gfx1250, wave32. FP32 GEMMs via V_WMMA_F32_16X16X4_F32,
// fragment-ready paired LDS layout (one ds_load_b64 per WMMA source),
// 128x64 block tile (32x32 per wave -> 1:1 DS-load:WMMA), double-buffered LDS.

typedef float v2f __attribute__((ext_vector_type(2)));
typedef float v8f __attribute__((ext_vector_type(8)));

#define TM 128
#define TN 64
#define TK 32
#define KP (TK / 2)        // 16 pair-rows per K chunk
#define AST (TM + 16)      // 144 float2 -> row stride 1152B == 128 (mod 256): half-wave bank split
#define BST (TN + 16)      // 80  float2 -> row stride 640B  == 128 (mod 256)

__device__ __forceinline__ v8f wmma_f32(v2f a, v2f b, v8f c) {
    return __builtin_amdgcn_wmma_f32_16x16x4_f32(false, a, false, b, (short)0, c, false, false);
}

// ---------------------------------------------------------------------------
// Kernel 1: per-row stats: xw = x_row . W ; deg1 = 1 + #(adj>0) ; rs = sum(adj)
// ---------------------------------------------------------------------------
__global__ __launch_bounds__(256) void row_stats_kernel(
    const float* __restrict__ x, const float* __restrict__ adj,
    const float* __restrict__ W, int N, int F,
    float* __restrict__ xw, float* __restrict__ dis,
    float* __restrict__ d2, float* __restrict__ rowflag)
{
    __shared__ float red[256];
    const int i = blockIdx.x, b = blockIdx.y, tid = threadIdx.x;
    const float* arow = adj + ((long long)b * N + i) * N;
    const float* xrow = x + ((long long)b * N + i) * F;

    float rs = 0.f, cnt = 0.f, xwp = 0.f;
    for (int j = tid; j < N; j += 256) {
        float a = arow[j];
        rs += a;
        cnt += (a > 0.f) ? 1.f : 0.f;
    }
    for (int f = tid; f < F; f += 256) xwp += xrow[f] * W[f];

    red[tid] = rs;  __syncthreads();
    for (int s = 128; s > 0; s >>= 1) { if (tid < s) red[tid] += red[tid + s]; __syncthreads(); }
    rs = red[0]; __syncthreads();
    red[tid] = cnt; __syncthreads();
    for (int s = 128; s > 0; s >>= 1) { if (tid < s) red[tid] += red[tid + s]; __syncthreads(); }
    cnt = red[0]; __syncthreads();
    red[tid] = xwp; __syncthreads();
    for (int s = 128; s > 0; s >>= 1) { if (tid < s) red[tid] += red[tid + s]; __syncthreads(); }
    xwp = red[0];

    if (tid == 0) {
        long long idx = (long long)b * N + i;
        xw[idx] = xwp;
        dis[idx] = rsqrtf(1.f + cnt);
        d2[idx] = rsqrtf(1.f + rs);
        rowflag[idx] = (rs > 0.f) ? 1.f : 0.f;
    }
}

// ---------------------------------------------------------------------------
// Kernel 2: alpha[b,i]
// ---------------------------------------------------------------------------
__global__ __launch_bounds__(256) void alpha_kernel(
    const float* __restrict__ adj, const float* __restrict__ xw,
    const float* __restrict__ dis, const float* __restrict__ bbias,
    const int* __restrict__ num_nodes, int N, float* __restrict__ alpha)
{
    __shared__ float red[256];
    const int i = blockIdx.x, b = blockIdx.y, tid = threadIdx.x;
    const float* arow = adj + ((long long)b * N + i) * N;
    const float* disb = dis + (long long)b * N;
    const float* xwb = xw + (long long)b * N;

    float acc = 0.f;
    for (int j = tid; j < N; j += 256) {
        float a = arow[j];
        if (a > 0.f) acc += disb[j] * xwb[j];
    }
    red[tid] = acc; __syncthreads();
    for (int s = 128; s > 0; s >>= 1) { if (tid < s) red[tid] += red[tid + s]; __syncthreads(); }

    if (tid == 0) {
        float di = disb[i];
        float p = di * (red[0] + di * xwb[i]) + bbias[0];
        float z = p * p;
        float sg = 1.f / (1.f + expf(-z));
        alpha[(long long)b * N + i] = (i < num_nodes[b]) ? sg : -1.f;
    }
}

// ---------------------------------------------------------------------------
// Kernel 3: cut_value[b] = k-th largest alpha among sentence nodes (id < ns)
// ---------------------------------------------------------------------------
__global__ __launch_bounds__(128) void cut_kernel(
    const float* __restrict__ alpha, const int* __restrict__ num_sentences,
    const int* __restrict__ nout, float* __restrict__ cutval, int N)
{
    __shared__ float vals[128];
    __shared__ float rv[128];
    __shared__ int   ri[128];
    const int b = blockIdx.x, tid = threadIdx.x;
    const int ns = num_sentences[b];
    vals[tid] = (tid < ns && tid < N) ? alpha[(long long)b * N + tid] : -1e30f;
    __syncthreads();

    const int k = nout[0];
    float cv = -1e30f;
    for (int it = 0; it < k; ++it) {
        rv[tid] = vals[tid]; ri[tid] = tid; __syncthreads();
        for (int s = 64; s > 0; s >>= 1) {
            if (tid < s && rv[tid + s] > rv[tid]) { rv[tid] = rv[tid + s]; ri[tid] = ri[tid + s]; }
            __syncthreads();
        }
        cv = rv[0];
        if (tid == 0) vals[ri[0]] = -1e30f;
        __syncthreads();
    }
    if (tid == 0) cutval[b] = cv;
}

// ---------------------------------------------------------------------------
// Kernel 4: cut_alpha = relu(alpha + 1e-7 - cv) ; index_mask = (alpha >= cv)
// ---------------------------------------------------------------------------
__global__ __launch_bounds__(256) void mask_kernel(
    const float* __restrict__ alpha, const float* __restrict__ cutval,
    int N, int total, float* __restrict__ cutalpha, float* __restrict__ mask_out)
{
    int idx = blockIdx.x * 256 + threadIdx.x;
    if (idx >= total) return;
    int b = idx / N;
    float a = alpha[idx];
    float cv = cutval[b];
    cutalpha[idx] = fmaxf(a + 1e-7f - cv, 0.f);
    mask_out[idx] = (a >= cv) ? 1.f : 0.f;
}

// ---------------------------------------------------------------------------
// Kernel 5: S row build + L1 normalize (assumes N <= 1024)
// ---------------------------------------------------------------------------
__global__ __launch_bounds__(256) void s_row_kernel(
    const float* __restrict__ adj, const float* __restrict__ d2,
    const float* __restrict__ rowflag, const float* __restrict__ cutalpha,
    int N, float* __restrict__ S)
{
    __shared__ float trow[1024];
    __shared__ float red[256];
    const int i = blockIdx.x, b = blockIdx.y, tid = threadIdx.x;
    const float* arow = adj + ((long long)b * N + i) * N;
    const float* d2b = d2 + (long long)b * N;
    const float* cab = cutalpha + (long long)b * N;
    const float fi = rowflag[(long long)b * N + i] * d2b[i];

    float sum = 0.f;
    for (int j = tid; j < N; j += 256) {
        float a = arow[j] + ((j == i) ? 1.f : 0.f);
        float t = fi * a * d2b[j] * cab[j];
        trow[j] = t;
        sum += t;
    }
    red[tid] = sum; __syncthreads();
    for (int s = 128; s > 0; s >>= 1) { if (tid < s) red[tid] += red[tid + s]; __syncthreads(); }
    float inv = 1.f / fmaxf(red[0], 1e-12f);
    float* srow = S + ((long long)b * N + i) * N;
    for (int j = tid; j < N; j += 256) srow[j] = trow[j] * inv;
}

// ---------------------------------------------------------------------------
// WMMA fp32 GEMM, C = A^T * B   (A stored [K][M], B stored [K][Ncols])
// 128x64 block tile, 8 waves, 32x32 per wave, double-buffered LDS,
// fragment-ready paired layout: LDS element = (val[k], val[k+1]).
// Requires M, Ncols, K even; M, Ncols multiples of 2 at tile edges (true here).
// ---------------------------------------------------------------------------
__global__ __launch_bounds__(256) void wmma_gemm_AtB(
    const float* __restrict__ Ab, const float* __restrict__ Bb, float* __restrict__ Cb,
    int M, int Ncols, int K, int lda, int ldb, int ldc,
    long long sA, long long sB, long long sC, int doFloor)
{
    __shared__ v2f As0[KP][AST], As1[KP][AST];
    __shared__ v2f Bs0[KP][BST], Bs1[KP][BST];

    const float* A = Ab + (long long)blockIdx.z * sA;
    const float* B = Bb + (long long)blockIdx.z * sB;
    float* C = Cb + (long long)blockIdx.z * sC;

    const int m0 = blockIdx.y * TM;
    const int n0 = blockIdx.x * TN;
    const int tid = threadIdx.x;
    const int wave = tid >> 5;
    const int lane = tid & 31;
    const int half = lane >> 4;
    const int l16 = lane & 15;
    const int am0 = (wave & 3) * 32;      // wave's 32-row region within tile
    const int bn0 = (wave >> 2) * 32;     // wave's 32-col region within tile

    v8f acc00 = {}, acc01 = {}, acc10 = {}, acc11 = {};

    float2 aR[4][2];
    float2 bR[2][2];
    const float2 z2 = make_float2(0.f, 0.f);
    const int numK = (K + TK - 1) / TK;

    // ---- prefetch chunk 0 into registers ----
#pragma unroll
    for (int p = 0; p < 4; ++p) {
        int it = p * 256 + tid;
        int kp = it >> 6, mmp = it & 63;
        int gk = 2 * kp, gm = m0 + 2 * mmp;
        bool ok = (gk < K) && (gm < M);
        aR[p][0] = ok ? *(const float2*)(A + (long long)gk * lda + gm) : z2;
        aR[p][1] = ok ? *(const float2*)(A + (long long)(gk + 1) * lda + gm) : z2;
    }
#pragma unroll
    for (int p = 0; p < 2; ++p) {
        int it = p * 256 + tid;
        int kp = it >> 5, nnp = it & 31;
        int gk = 2 * kp, gn = n0 + 2 * nnp;
        bool ok = (gk < K) && (gn < Ncols);
        bR[p][0] = ok ? *(const float2*)(B + (long long)gk * ldb + gn) : z2;
        bR[p][1] = ok ? *(const float2*)(B + (long long)(gk + 1) * ldb + gn) : z2;
    }
    // ---- commit chunk 0 to buffer 0 ----
#pragma unroll
    for (int p = 0; p < 4; ++p) {
        int it = p * 256 + tid;
        int kp = it >> 6, mmp = it & 63;
        v2f t0 = { aR[p][0].x, aR[p][1].x };
        v2f t1 = { aR[p][0].y, aR[p][1].y };
        As0[kp][2 * mmp] = t0; As0[kp][2 * mmp + 1] = t1;
    }
#pragma unroll
    for (int p = 0; p < 2; ++p) {
        int it = p * 256 + tid;
        int kp = it >> 5, nnp = it & 31;
        v2f t0 = { bR[p][0].x, bR[p][1].x };
        v2f t1 = { bR[p][0].y, bR[p][1].y };
        Bs0[kp][2 * nnp] = t0; Bs0[kp][2 * nnp + 1] = t1;
    }
    __syncthreads();

    for (int c = 0; c < numK; ++c) {
        // prefetch next chunk while computing this one
        if (c + 1 < numK) {
            int k0 = (c + 1) * TK;
#pragma unroll
            for (int p = 0; p < 4; ++p) {
                int it = p * 256 + tid;
                int kp = it >> 6, mmp = it & 63;
                int gk = k0 + 2 * kp, gm = m0 + 2 * mmp;
                bool ok = (gk < K) && (gm < M);
                aR[p][0] = ok ? *(const float2*)(A + (long long)gk * lda + gm) : z2;
                aR[p][1] = ok ? *(const float2*)(A + (long long)(gk + 1) * lda + gm) : z2;
            }
#pragma unroll
            for (int p = 0; p < 2; ++p) {
                int it = p * 256 + tid;
                int kp = it >> 5, nnp = it & 31;
                int gk = k0 + 2 * kp, gn = n0 + 2 * nnp;
                bool ok = (gk < K) && (gn < Ncols);
                bR[p][0] = ok ? *(const float2*)(B + (long long)gk * ldb + gn) : z2;
                bR[p][1] = ok ? *(const float2*)(B + (long long)(gk + 1) * ldb + gn) : z2;
            }
        }

        const v2f (*Ac)[AST] = (c & 1) ? As1 : As0;
        const v2f (*Bc)[BST] = (c & 1) ? Bs1 : Bs0;
#pragma unroll
        for (int kk = 0; kk < KP; kk += 2) {        // kp = kk + half -> K slice 2*kk..2*kk+3
            v2f a0 = Ac[kk + half][am0 + l16];
            v2f a1 = Ac[kk + half][am0 + 16 + l16];
            v2f b0 = Bc[kk + half][bn0 + l16];
            v2f b1 = Bc[kk + half][bn0 + 16 + l16];
            acc00 = wmma_f32(a0, b0, acc00);
            acc01 = wmma_f32(a0, b1, acc01);
            acc10 = wmma_f32(a1, b0, acc10);
            acc11 = wmma_f32(a1, b1, acc11);
        }
        __syncthreads();

        if (c + 1 < numK) {
            v2f (*Ad)[AST] = (c & 1) ? As0 : As1;
            v2f (*Bd)[BST] = (c & 1) ? Bs0 : Bs1;
#pragma unroll
            for (int p = 0; p < 4; ++p) {
                int it = p * 256 + tid;
                int kp = it >> 6, mmp = it & 63;
                v2f t0 = { aR[p][0].x, aR[p][1].x };
                v2f t1 = { aR[p][0].y, aR[p][1].y };
                Ad[kp][2 * mmp] = t0; Ad[kp][2 * mmp + 1] = t1;
            }
#pragma unroll
            for (int p = 0; p < 2; ++p) {
                int it = p * 256 + tid;
                int kp = it >> 5, nnp = it & 31;
                v2f t0 = { bR[p][0].x, bR[p][1].x };
                v2f t1 = { bR[p][0].y, bR[p][1].y };
                Bd[kp][2 * nnp] = t0; Bd[kp][2 * nnp + 1] = t1;
            }
        }
        __syncthreads();
    }

    // ---- epilogue ----
#pragma unroll
    for (int r = 0; r < 8; ++r) {
        int row0 = m0 + am0 + half * 8 + r;
        int row1 = row0 + 16;
        int c0 = n0 + bn0 + l16;
        int c1 = c0 + 16;
        float v00 = acc00[r], v01 = acc01[r], v10 = acc10[r], v11 = acc11[r];
        if (doFloor) {
            v00 = floorf(v00 * 1e4f) / 1e4f; v01 = floorf(v01 * 1e4f) / 1e4f;
            v10 = floorf(v10 * 1e4f) / 1e4f; v11 = floorf(v11 * 1e4f) / 1e4f;
        }
        if (row0 < M) {
            if (c0 < Ncols) C[(long long)row0 * ldc + c0] = v00;
            if (c1 < Ncols) C[(long long)row0 * ldc + c1] = v01;
        }
        if (row1 < M) {
            if (c0 < Ncols) C[(long long)row1 * ldc + c0] = v10;
            if (c1 < Ncols) C[(long long)row1 * ldc + c1] = v11;
        }
    }
}

// ---------------------------------------------------------------------------
// WMMA fp32 GEMM, C = A * B   (A stored [M][K], B stored [K][Ncols])
// Same tile/fragment scheme; A staged via float4 rows (K multiple of 4).
// ---------------------------------------------------------------------------
__global__ __launch_bounds__(256) void wmma_gemm_AB(
    const float* __restrict__ Ab, const float* __restrict__ Bb, float* __restrict__ Cb,
    int M, int Ncols, int K, int lda, int ldb, int ldc,
    long long sA, long long sB, long long sC, int doFloor)
{
    __shared__ v2f As0[KP][AST], As1[KP][AST];
    __shared__ v2f Bs0[KP][BST], Bs1[KP][BST];

    const float* A = Ab + (long long)blockIdx.z * sA;
    const float* B = Bb + (long long)blockIdx.z * sB;
    float* C = Cb + (long long)blockIdx.z * sC;

    const int m0 = blockIdx.y * TM;
    const int n0 = blockIdx.x * TN;
    const int tid = threadIdx.x;
    const int wave = tid >> 5;
    const int lane = tid & 31;
    const int half = lane >> 4;
    const int l16 = lane & 15;
    const int am0 = (wave & 3) * 32;
    const int bn0 = (wave >> 2) * 32;

    v8f acc00 = {}, acc01 = {}, acc10 = {}, acc11 = {};

    float4 aR4[4];
    float2 bR[2][2];
    const float2 z2 = make_float2(0.f, 0.f);
    const float4 z4 = make_float4(0.f, 0.f, 0.f, 0.f);
    const int numK = (K + TK - 1) / TK;

#pragma unroll
    for (int p = 0; p < 4; ++p) {
        int it = p * 256 + tid;
        int mm = it >> 3, t4 = it & 7;
        int gm = m0 + mm, gk = 4 * t4;
        aR4[p] = (gm < M && gk < K) ? *(const float4*)(A + (long long)gm * lda + gk) : z4;
    }
#pragma unroll
    for (int p = 0; p < 2; ++p) {
        int it = p * 256 + tid;
        int kp = it >> 5, nnp = it & 31;
        int gk = 2 * kp, gn = n0 + 2 * nnp;
        bool ok = (gk < K) && (gn < Ncols);
        bR[p][0] = ok ? *(const float2*)(B + (long long)gk * ldb + gn) : z2;
        bR[p][1] = ok ? *(const float2*)(B + (long long)(gk + 1) * ldb + gn) : z2;
    }
#pragma unroll
    for (int p = 0; p < 4; ++p) {
        int it = p * 256 + tid;
        int mm = it >> 3, t4 = it & 7;
        v2f t0 = { aR4[p].x, aR4[p].y };
        v2f t1 = { aR4[p].z, aR4[p].w };
        As0[2 * t4][mm] = t0; As0[2 * t4 + 1][mm] = t1;
    }
#pragma unroll
    for (int p = 0; p < 2; ++p) {
        int it = p * 256 + tid;
        int kp = it >> 5, nnp = it & 31;
        v2f t0 = { bR[p][0].x, bR[p][1].x };
        v2f t1 = { bR[p][0].y, bR[p][1].y };
        Bs0[kp][2 * nnp] = t0; Bs0[kp][2 * nnp + 1] = t1;
    }
    __syncthreads();

    for (int c = 0; c < numK; ++c) {
        if (c + 1 < numK) {
            int k0 = (c + 1) * TK;
#pragma unroll
            for (int p = 0; p < 4; ++p) {
                int it = p * 256 + tid;
                int mm = it >> 3, t4 = it & 7;
                int gm = m0 + mm, gk = k0 + 4 * t4;
                aR4[p] = (gm < M && gk < K) ? *(const float4*)(A + (long long)gm * lda + gk) : z4;
            }
#pragma unroll
            for (int p = 0; p < 2; ++p) {
                int it = p * 256 + tid;
                int kp = it >> 5, nnp = it & 31;
                int gk = k0 + 2 * kp, gn = n0 + 2 * nnp;
                bool ok = (gk < K) && (gn < Ncols);
                bR[p][0] = ok ? *(const float2*)(B + (long long)gk * ldb + gn) : z2;
                bR[p][1] = ok ? *(const float2*)(B + (long long)(gk + 1) * ldb + gn) : z2;
            }
        }

        const v2f (*Ac)[AST] = (c & 1) ? As1 : As0;
        const v2f (*Bc)[BST] = (c & 1) ? Bs1 : Bs0;
#pragma unroll
        for (int kk = 0; kk < KP; kk += 2) {
            v2f a0 = Ac[kk + half][am0 + l16];
            v2f a1 = Ac[kk + half][am0 + 16 + l16];
            v2f b0 = Bc[kk + half][bn0 + l16];
            v2f b1 = Bc[kk + half][bn0 + 16 + l16];
            acc00 = wmma_f32(a0, b0, acc00);
            acc01 = wmma_f32(a0, b1, acc01);
            acc10 = wmma_f32(a1, b0, acc10);
            acc11 = wmma_f32(a1, b1, acc11);
        }
        __syncthreads();

        if (c + 1 < numK) {
            v2f (*Ad)[AST] = (c & 1) ? As0 : As1;
            v2f (*Bd)[BST] = (c & 1) ? Bs0 : Bs1;
#pragma unroll
            for (int p = 0; p < 4; ++p) {
                int it = p * 256 + tid;
                int mm = it >> 3, t4 = it & 7;
                v2f t0 = { aR4[p].x, aR4[p].y };
                v2f t1 = { aR4[p].z, aR4[p].w };
                Ad[2 * t4][mm] = t0; Ad[2 * t4 + 1][mm] = t1;
            }
#pragma unroll
            for (int p = 0; p < 2; ++p) {
                int it = p * 256 + tid;
                int kp = it >> 5, nnp = it & 31;
                v2f t0 = { bR[p][0].x, bR[p][1].x };
                v2f t1 = { bR[p][0].y, bR[p][1].y };
                Bd[kp][2 * nnp] = t0; Bd[kp][2 * nnp + 1] = t1;
            }
        }
        __syncthreads();
    }

#pragma unroll
    for (int r = 0; r < 8; ++r) {
        int row0 = m0 + am0 + half * 8 + r;
        int row1 = row0 + 16;
        int c0 = n0 + bn0 + l16;
        int c1 = c0 + 16;
        float v00 = acc00[r], v01 = acc01[r], v10 = acc10[r], v11 = acc11[r];
        if (doFloor) {
            v00 = floorf(v00 * 1e4f) / 1e4f; v01 = floorf(v01 * 1e4f) / 1e4f;
            v10 = floorf(v10 * 1e4f) / 1e4f; v11 = floorf(v11 * 1e4f) / 1e4f;
        }
        if (row0 < M) {
            if (c0 < Ncols) C[(long long)row0 * ldc + c0] = v00;
            if (c1 < Ncols) C[(long long)row0 * ldc + c1] = v01;
        }
        if (row1 < M) {
            if (c0 < Ncols) C[(long long)row1 * ldc + c0] = v10;
            if (c1 < Ncols) C[(long long)row1 * ldc + c1] = v11;
        }
    }
}

// ---------------------------------------------------------------------------
extern "C" void kernel_launch(void* const* d_in, const int* in_sizes, int n_in,
                              void* d_out, int out_size, void* d_ws, size_t ws_size,
                              hipStream_t stream) {
    const float* x     = (const float*)d_in[0];   // [B,N,F]
    const float* adj   = (const float*)d_in[1];   // [B,N,N]
    const float* W     = (const float*)d_in[2];   // [F,1]
    const float* bbias = (const float*)d_in[3];   // [1]
    const int* num_nodes = (const int*)d_in[4];   // [B]
    const int* num_sent  = (const int*)d_in[5];   // [B]
    const int* nout      = (const int*)d_in[6];   // scalar

    const int F = in_sizes[2];
    const int B = in_sizes[4];
    const int N = (int)((long long)in_sizes[0] / ((long long)B * F));

    float* out = (float*)d_out;
    float* xc_out     = out;                                        // [B,N,F]
    float* coarse_out = xc_out + (long long)B * N * F;              // [B,N,N]
    float* S_out      = coarse_out + (long long)B * N * N;          // [B,N,N]
    float* mask_out   = S_out + (long long)B * N * N;               // [B,N]

    const long long BN = (long long)B * N;
    float* ws = (float*)d_ws;
    float* xw       = ws; ws += BN;
    float* dis      = ws; ws += BN;
    float* d2       = ws; ws += BN;
    float* rowflag  = ws; ws += BN;
    float* alpha    = ws; ws += BN;
    float* cutalpha = ws; ws += BN;
    float* cutval   = ws; ws += B;
    uintptr_t p = (uintptr_t)ws;
    p = (p + 255) & ~(uintptr_t)255;
    float* Tbuf = (float*)p;                                        // [N,N]

    dim3 blk(256);

    row_stats_kernel<<<dim3(N, B), blk, 0, stream>>>(x, adj, W, N, F, xw, dis, d2, rowflag);
    alpha_kernel<<<dim3(N, B), blk, 0, stream>>>(adj, xw, dis, bbias, num_nodes, N, alpha);
    cut_kernel<<<dim3(B), dim3(128), 0, stream>>>(alpha, num_sent, nout, cutval, N);
    int total = (int)BN;
    mask_kernel<<<dim3((total + 255) / 256), blk, 0, stream>>>(alpha, cutval, N, total, cutalpha, mask_out);
    s_row_kernel<<<dim3(N, B), blk, 0, stream>>>(adj, d2, rowflag, cutalpha, N, S_out);

    // x_c = S^T x : all batches in one launch
    dim3 gxc((F + TN - 1) / TN, (N + TM - 1) / TM, B);
    wmma_gemm_AtB<<<gxc, blk, 0, stream>>>(S_out, x, xc_out,
        N, F, N, N, F, F,
        (long long)N * N, (long long)N * F, (long long)N * F, 0);

    // coarse = (S^T adj) S per batch (4MB T scratch reused; stream serializes)
    dim3 g2((N + TN - 1) / TN, (N + TM - 1) / TM, 1);
    for (int b = 0; b < B; ++b) {
        const float* Sb = S_out + (long long)b * N * N;
        const float* Adjb = adj + (long long)b * N * N;
        wmma_gemm_AtB<<<g2, blk, 0, stream>>>(Sb, Adjb, Tbuf,
            N, N, N, N, N, N, 0, 0, 0, 0);
        wmma_gemm_AB<<<g2, blk, 0, stream>>>(Tbuf, Sb, coarse_out + (long long)b * N * N,
            N, N, N, N, N, N, 0, 0, 0, 1);
    }
}